// RFEncoder_15848429322973
// MI455X (gfx1250) — compile-verified
//
#include <hip/hip_runtime.h>

// ---------------------------------------------------------------------------
// GATv2 hetero 2-layer network for MI455X (gfx1250, wave32).
// Layer-2 node transforms use V_WMMA_F32_16X16X4_F32 (fp32 matrix core path,
// matches the fp32 reference numerics). Edge phase = 3-pass segment softmax
// with L2-resident atomics.
// ---------------------------------------------------------------------------

typedef __attribute__((ext_vector_type(2))) float v2f;
typedef __attribute__((ext_vector_type(8))) float v8f;

#define NEG_SLOPE 0.2f

__device__ __forceinline__ float lrelu(float v) {
    return fmaxf(v, 0.0f) + NEG_SLOPE * fminf(v, 0.0f);
}

// ordered-int trick atomic max for fp32 (no NaNs expected)
__device__ __forceinline__ void atomicMaxF(float* addr, float val) {
    if (val >= 0.0f)
        atomicMax((int*)addr, __float_as_int(val));
    else
        atomicMin((unsigned int*)addr, __float_as_uint(val));
}

// ---------------------------------------------------------------------------
__global__ void fill_kernel(float* __restrict__ p, float v, int n) {
    int i = blockIdx.x * blockDim.x + threadIdx.x;
    if (i < n) p[i] = v;
}

// ---------------------------------------------------------------------------
// Layer 1 transform: x[N,2] @ W[2,128] + b  -> out[N,128].
// K=2: scalar FMA. 32 threads per node, 4 channels each (float4).
// blockIdx.y: 0 => (Wl,bl)->xl ; 1 => (Wr,br)->xr
__global__ void gemm_l1(const float* __restrict__ x,
                        const float* __restrict__ Wl, const float* __restrict__ bl,
                        float* __restrict__ xl,
                        const float* __restrict__ Wr, const float* __restrict__ br,
                        float* __restrict__ xr, int n) {
    int tid = blockIdx.x * blockDim.x + threadIdx.x;
    int node = tid >> 5;
    int q = tid & 31;
    if (node >= n) return;
    const float* W = blockIdx.y ? Wr : Wl;
    const float* b = blockIdx.y ? br : bl;
    float* out = blockIdx.y ? xr : xl;
    float x0 = x[node * 2 + 0];
    float x1 = x[node * 2 + 1];
    int c0 = q * 4;
    float4 w0 = *(const float4*)(W + c0);
    float4 w1 = *(const float4*)(W + 128 + c0);
    float4 bb = *(const float4*)(b + c0);
    float4 o;
    o.x = fmaf(x0, w0.x, fmaf(x1, w1.x, bb.x));
    o.y = fmaf(x0, w0.y, fmaf(x1, w1.y, bb.y));
    o.z = fmaf(x0, w0.z, fmaf(x1, w1.z, bb.z));
    o.w = fmaf(x0, w0.w, fmaf(x1, w1.w, bb.w));
    *(float4*)(out + (size_t)node * 128 + c0) = o;
}

// ---------------------------------------------------------------------------
// Layer 2 transform via WMMA: h[N,64] @ W[64,128] + b -> out[N,128].
// One wave per 16x16 output tile; 8 waves/block cover all 128 columns of a
// 16-row stripe. K=64 in 16 steps of V_WMMA_F32_16X16X4_F32.
// fp32 A 16x4 layout: lane L holds M=L%16, K = {0,1} (L<16) or {2,3} (L>=16).
// fp32 B 4x16 layout: lane L holds N=L%16, K = {0,1} (L<16) or {2,3} (L>=16).
// fp32 C 16x16 layout: vgpr i -> row = i + 8*(L/16), col = L%16.
__global__ void gemm_l2_wmma(const float* __restrict__ h,
                             const float* __restrict__ Wl, const float* __restrict__ bl,
                             float* __restrict__ xl,
                             const float* __restrict__ Wr, const float* __restrict__ br,
                             float* __restrict__ xr, int n) {
    const float* W = blockIdx.y ? Wr : Wl;
    const float* b = blockIdx.y ? br : bl;
    float* out = blockIdx.y ? xr : xl;

    int wave = threadIdx.x >> 5;       // 0..7 -> column tile
    int lane = threadIdx.x & 31;
    int row0 = blockIdx.x * 16;
    int col0 = wave * 16;
    int mrow = lane & 15;
    int hi = lane >> 4;                // 0 or 1 (selects K pair)

    int arow_idx = row0 + mrow;
    if (arow_idx >= n) arow_idx = n - 1;     // clamp (M divides evenly anyway)
    const float* arow = h + (size_t)arow_idx * 64 + 2 * hi;
    const float* bcol = W + (size_t)(2 * hi) * 128 + col0 + mrow;

    v8f c = {};
    #pragma unroll
    for (int k = 0; k < 64; k += 4) {
        v2f a = *(const v2f*)(arow + k);       // K = k+2*hi, k+2*hi+1 (contiguous)
        v2f bf;
        bf.x = bcol[(size_t)k * 128];          // K = k+2*hi   row of W
        bf.y = bcol[(size_t)(k + 1) * 128];    // K = k+2*hi+1 row of W
        c = __builtin_amdgcn_wmma_f32_16x16x4_f32(
                false, a, false, bf, (short)0, c, false, false);
    }

    int colg = col0 + mrow;
    float bias = b[colg];
    #pragma unroll
    for (int i = 0; i < 8; ++i) {
        int row = row0 + i + 8 * hi;
        if (row < n)
            out[(size_t)row * 128 + colg] = c[i] + bias;
    }
}

// ---------------------------------------------------------------------------
// Pass A: per-edge attention logits + segment max.
// score_h = sum_c lrelu(xl[src,h,c] + xr[dst,h,c]) * att[h,c]
__global__ void edge_score(const float* __restrict__ xl, const float* __restrict__ xr,
                           const int* __restrict__ src, const int* __restrict__ dst,
                           const float* __restrict__ att,
                           float* __restrict__ sc, float* __restrict__ m, int E) {
    int e = blockIdx.x * blockDim.x + threadIdx.x;
    if (e >= E) return;
    int s = src[e], d = dst[e];
    const float* pl = xl + (size_t)s * 128;
    const float* pr = xr + (size_t)d * 128;
    float s0 = 0.0f, s1 = 0.0f;
    #pragma unroll 4
    for (int c = 0; c < 64; c += 4) {
        float4 a = *(const float4*)(pl + c);
        float4 r = *(const float4*)(pr + c);
        float4 w = *(const float4*)(att + c);
        s0 += lrelu(a.x + r.x) * w.x + lrelu(a.y + r.y) * w.y +
              lrelu(a.z + r.z) * w.z + lrelu(a.w + r.w) * w.w;
    }
    #pragma unroll 4
    for (int c = 64; c < 128; c += 4) {
        float4 a = *(const float4*)(pl + c);
        float4 r = *(const float4*)(pr + c);
        float4 w = *(const float4*)(att + c);
        s1 += lrelu(a.x + r.x) * w.x + lrelu(a.y + r.y) * w.y +
              lrelu(a.z + r.z) * w.z + lrelu(a.w + r.w) * w.w;
    }
    sc[(size_t)e * 2 + 0] = s0;
    sc[(size_t)e * 2 + 1] = s1;
    atomicMaxF(m + (size_t)d * 2 + 0, s0);
    atomicMaxF(m + (size_t)d * 2 + 1, s1);
}

// Pass B: ex = exp(score - m[dst]); denom[dst] += ex (per head).
__global__ void edge_exp(const int* __restrict__ dst,
                         float* __restrict__ sc, const float* __restrict__ m,
                         float* __restrict__ den, int E) {
    int e = blockIdx.x * blockDim.x + threadIdx.x;
    if (e >= E) return;
    int d = dst[e];
    float e0 = __expf(sc[(size_t)e * 2 + 0] - m[(size_t)d * 2 + 0]);
    float e1 = __expf(sc[(size_t)e * 2 + 1] - m[(size_t)d * 2 + 1]);
    sc[(size_t)e * 2 + 0] = e0;
    sc[(size_t)e * 2 + 1] = e1;
    atomicAdd(den + (size_t)d * 2 + 0, e0);
    atomicAdd(den + (size_t)d * 2 + 1, e1);
}

// Pass C: acc[dst, c%64] += 0.25 * alpha_h * xl[src, c]  (head-mean 0.5 *
// hetero-mean 0.5 folded in). 32 lanes per edge, 4 channels per lane.
__global__ void edge_accum(const float* __restrict__ xl,
                           const int* __restrict__ src, const int* __restrict__ dst,
                           const float* __restrict__ sc, const float* __restrict__ den,
                           float* __restrict__ acc, int E) {
    int tid = blockIdx.x * blockDim.x + threadIdx.x;
    int e = tid >> 5;
    if (e >= E) return;
    int q = tid & 31;
    int s = src[e], d = dst[e];
    int h = q >> 4;                       // head for these 4 channels
    float denom = den[(size_t)d * 2 + h] + 1e-16f;
    float alpha = sc[(size_t)e * 2 + h] / denom;
    float scale = 0.25f * alpha;
    int c0 = q * 4;
    float4 v = *(const float4*)(xl + (size_t)s * 128 + c0);
    float* pa = acc + (size_t)d * 64 + (c0 & 63);
    atomicAdd(pa + 0, v.x * scale);
    atomicAdd(pa + 1, v.y * scale);
    atomicAdd(pa + 2, v.z * scale);
    atomicAdd(pa + 3, v.w * scale);
}

// out[n,c] = (relu?)( acc[n,c] + 0.5*(bias_a[c] + bias_b[c]) )
__global__ void finalize_kernel(const float* __restrict__ acc,
                                const float* __restrict__ bias_a,
                                const float* __restrict__ bias_b,
                                float* __restrict__ out, int total, int do_relu) {
    int i = blockIdx.x * blockDim.x + threadIdx.x;
    if (i >= total) return;
    int c = i & 63;
    float v = acc[i] + 0.5f * (bias_a[c] + bias_b[c]);
    if (do_relu) v = fmaxf(v, 0.0f);
    out[i] = v;
}

// ---------------------------------------------------------------------------
extern "C" void kernel_launch(void* const* d_in, const int* in_sizes, int n_in,
                              void* d_out, int out_size, void* d_ws, size_t ws_size,
                              hipStream_t stream) {
    const float* x      = (const float*)d_in[0];
    const int*   ea     = (const int*)d_in[1];
    const int*   er     = (const int*)d_in[2];
    const float* l1_Wl  = (const float*)d_in[3];
    const float* l1_bl  = (const float*)d_in[4];
    const float* l1_Wr  = (const float*)d_in[5];
    const float* l1_br  = (const float*)d_in[6];
    const float* l1_att = (const float*)d_in[7];
    const float* l1_bias= (const float*)d_in[8];
    const float* l2_Wl  = (const float*)d_in[9];
    const float* l2_bl  = (const float*)d_in[10];
    const float* l2_Wr  = (const float*)d_in[11];
    const float* l2_br  = (const float*)d_in[12];
    const float* l2_att = (const float*)d_in[13];
    const float* l2_bias= (const float*)d_in[14];

    const int N  = in_sizes[0] / 2;     // IN_DIM = 2
    const int Ea = in_sizes[1] / 2;
    const int Er = in_sizes[2] / 2;
    const int Emax = Ea > Er ? Ea : Er;

    float* acc = (float*)d_ws;                    // N*64  (layer-1 acc -> h)
    float* xlb = acc + (size_t)N * 64;            // N*128
    float* xrb = xlb + (size_t)N * 128;           // N*128
    float* scb = xrb + (size_t)N * 128;           // Emax*2
    float* mb  = scb + (size_t)Emax * 2;          // N*2
    float* db  = mb  + (size_t)N * 2;             // N*2
    float* outp = (float*)d_out;                  // N*64 (layer-2 acc + result)

    const int T = 256;
    const float NEG_INF = -__builtin_huge_valf();
    auto cdiv = [](int a, int b) { return (a + b - 1) / b; };

    // ---------------- layer 1 ----------------
    fill_kernel<<<cdiv(N * 64, T), T, 0, stream>>>(acc, 0.0f, N * 64);
    for (int rel = 0; rel < 2; ++rel) {
        int E = (rel == 0) ? Ea : Er;
        const int* src = (rel == 0) ? ea : er;
        const int* dst = src + E;
        fill_kernel<<<cdiv(N * 2, T), T, 0, stream>>>(mb, NEG_INF, N * 2);
        fill_kernel<<<cdiv(N * 2, T), T, 0, stream>>>(db, 0.0f, N * 2);
        gemm_l1<<<dim3(cdiv(N * 32, T), 2), T, 0, stream>>>(
            x, l1_Wl + rel * 256, l1_bl + rel * 128, xlb,
               l1_Wr + rel * 256, l1_br + rel * 128, xrb, N);
        edge_score<<<cdiv(E, T), T, 0, stream>>>(xlb, xrb, src, dst,
                                                 l1_att + rel * 128, scb, mb, E);
        edge_exp<<<cdiv(E, T), T, 0, stream>>>(dst, scb, mb, db, E);
        edge_accum<<<cdiv(E * 32, T), T, 0, stream>>>(xlb, src, dst, scb, db, acc, E);
    }
    // in-place: acc becomes h = relu(acc + 0.5*(bias_adj + bias_ray))
    finalize_kernel<<<cdiv(N * 64, T), T, 0, stream>>>(acc, l1_bias, l1_bias + 64,
                                                       acc, N * 64, 1);

    // ---------------- layer 2 ----------------
    fill_kernel<<<cdiv(N * 64, T), T, 0, stream>>>(outp, 0.0f, N * 64);
    for (int rel = 0; rel < 2; ++rel) {
        int E = (rel == 0) ? Ea : Er;
        const int* src = (rel == 0) ? ea : er;
        const int* dst = src + E;
        fill_kernel<<<cdiv(N * 2, T), T, 0, stream>>>(mb, NEG_INF, N * 2);
        fill_kernel<<<cdiv(N * 2, T), T, 0, stream>>>(db, 0.0f, N * 2);
        gemm_l2_wmma<<<dim3(cdiv(N, 16), 2), T, 0, stream>>>(
            acc, l2_Wl + rel * 64 * 128, l2_bl + rel * 128, xlb,
                 l2_Wr + rel * 64 * 128, l2_br + rel * 128, xrb, N);
        edge_score<<<cdiv(E, T), T, 0, stream>>>(xlb, xrb, src, dst,
                                                 l2_att + rel * 128, scb, mb, E);
        edge_exp<<<cdiv(E, T), T, 0, stream>>>(dst, scb, mb, db, E);
        edge_accum<<<cdiv(E * 32, T), T, 0, stream>>>(xlb, src, dst, scb, db, outp, E);
    }
    finalize_kernel<<<cdiv(N * 64, T), T, 0, stream>>>(outp, l2_bias, l2_bias + 64,
                                                       outp, N * 64, 0);
}